// RecallAtKSurrogateLoss_88364657148359
// MI455X (gfx1250) — compile-verified
//
#include <hip/hip_runtime.h>
#include <math.h>

// ---------------------------------------------------------------------------
// RecallAtK surrogate loss for MI455X (gfx1250, wave32, WMMA, LDS-staged).
// sim = E * E^T computed tile-wise with V_WMMA_F32_16X16X4_F32 fed from LDS.
// Two passes: (1) max over negatives, (2) sigmoid-sum over positives.
// ---------------------------------------------------------------------------

typedef __attribute__((ext_vector_type(2))) float v2f;
typedef __attribute__((ext_vector_type(8))) float v8f;

#define B_SIZE   8192
#define D_SIZE   128
#define NCLS     1000
#define BM       128         // rows of sim per block (4 waves x 32 rows)
#define BN       64          // cols of sim per block
#define LDP      132         // padded LDS row stride (floats): 132 mod 64 == 4
#define NEG_INF  (-3.0e38f)

// Monotonic float max via sign-split integer atomics (portable, no CAS loop).
__device__ __forceinline__ void atomicMaxF(float* addr, float val) {
    if (val >= 0.0f) {
        atomicMax((int*)addr, __float_as_int(val));
    } else {
        atomicMin((unsigned int*)addr, (unsigned int)__float_as_int(val));
    }
}

// Coalesced float4 staging of the row-panel (BM x 128) and col-panel (BN x 128)
// of E into padded LDS.
__device__ __forceinline__ void stage_tiles(const float* __restrict__ E,
                                            int row0, int col0, int tid,
                                            float* sA, float* sB) {
#pragma unroll
    for (int idx = tid; idx < BM * 32; idx += 128) {
        int r = idx >> 5, c4 = (idx & 31) << 2;
        *(float4*)(sA + r * LDP + c4) =
            *(const float4*)(E + (size_t)(row0 + r) * D_SIZE + c4);
    }
#pragma unroll
    for (int idx = tid; idx < BN * 32; idx += 128) {
        int r = idx >> 5, c4 = (idx & 31) << 2;
        *(float4*)(sB + r * LDP + c4) =
            *(const float4*)(E + (size_t)(col0 + r) * D_SIZE + c4);
    }
}

// One wave computes a 32x64 strip of sim: 2x4 accumulator tiles of 16x16.
// A-frag (16x4 f32): lane M = lane&15, VGPRs hold K = kh, kh+1, kh = 2*(lane>>4).
// B-frag (4x16) mirrors it with N = lane&15 (B = E^T, so B[k][n] = E[col+n][k]).
__device__ __forceinline__ void wave_gemm_lds(const float* __restrict__ sA,
                                              const float* __restrict__ sB,
                                              int wid, int lane, v8f acc[2][4]) {
    const int m  = lane & 15;
    const int kh = (lane >> 4) << 1;      // 0 or 2
    const float* a0 = sA + (32 * wid +      m) * LDP + kh;
    const float* a1 = sA + (32 * wid + 16 + m) * LDP + kh;
#pragma unroll
    for (int rt = 0; rt < 2; ++rt)
#pragma unroll
        for (int t = 0; t < 4; ++t)
            acc[rt][t] = (v8f){0.f, 0.f, 0.f, 0.f, 0.f, 0.f, 0.f, 0.f};

#pragma unroll 4
    for (int kk = 0; kk < D_SIZE; kk += 4) {
        v2f aF0 = *(const v2f*)(a0 + kk);
        v2f aF1 = *(const v2f*)(a1 + kk);
#pragma unroll
        for (int t = 0; t < 4; ++t) {
            v2f bF = *(const v2f*)(sB + (16 * t + m) * LDP + kh + kk);
            acc[0][t] = __builtin_amdgcn_wmma_f32_16x16x4_f32(
                false, aF0, false, bF, (short)0, acc[0][t], false, false);
            acc[1][t] = __builtin_amdgcn_wmma_f32_16x16x4_f32(
                false, aF1, false, bF, (short)0, acc[1][t], false, false);
        }
    }
}

__global__ void rk_init_kernel(float* maxneg, float* possum, int* cnt, float* out) {
    int i = blockIdx.x * blockDim.x + threadIdx.x;
    if (i < B_SIZE) { maxneg[i] = NEG_INF; possum[i] = 0.0f; }
    if (i < NCLS)   cnt[i] = 0;
    if (i == 0)     out[0] = 0.0f;
}

__global__ void rk_hist_kernel(const int* __restrict__ labels, int* __restrict__ cnt) {
    int i = blockIdx.x * blockDim.x + threadIdx.x;
    if (i < B_SIZE) atomicAdd(&cnt[labels[i]], 1);
}

// Pass 1: max over negatives per row.
__global__ void __launch_bounds__(128)
rk_maxneg_kernel(const float* __restrict__ E,
                 const int* __restrict__ labels,
                 float* __restrict__ maxneg) {
    __shared__ float sA[BM * LDP];
    __shared__ float sB[BN * LDP];
    __shared__ int   rlab[BM];
    __shared__ int   clab[BN];

    const int row0 = blockIdx.y * BM;
    const int col0 = blockIdx.x * BN;
    const int tid  = threadIdx.x;

    rlab[tid] = labels[row0 + tid];
    if (tid < BN) clab[tid] = labels[col0 + tid];
    stage_tiles(E, row0, col0, tid, sA, sB);
    __syncthreads();

    const int wid  = tid >> 5;
    const int lane = tid & 31;
    const int half = lane >> 4;
    const int n16  = lane & 15;

    v8f acc[2][4];
    wave_gemm_lds(sA, sB, wid, lane, acc);

#pragma unroll
    for (int rt = 0; rt < 2; ++rt) {
#pragma unroll
        for (int r = 0; r < 8; ++r) {
            const int rloc = 32 * wid + 16 * rt + r + 8 * half;  // M = r + 8*half
            const int rl   = rlab[rloc];
            float mx = NEG_INF;
#pragma unroll
            for (int t = 0; t < 4; ++t) {
                float v = acc[rt][t][r];
                mx = fmaxf(mx, (rl != clab[16 * t + n16]) ? v : NEG_INF);
            }
            // reduce across the 16 lanes of this half-wave (rows stay separate)
#pragma unroll
            for (int off = 1; off < 16; off <<= 1)
                mx = fmaxf(mx, __shfl_xor(mx, off, 32));
            if (n16 == 0) atomicMaxF(&maxneg[row0 + rloc], mx);
        }
    }
}

// Pass 2: sum of sigmoid(max_neg[row] - sim) over positives per row.
__global__ void __launch_bounds__(128)
rk_possum_kernel(const float* __restrict__ E,
                 const int* __restrict__ labels,
                 const float* __restrict__ maxneg,
                 float* __restrict__ possum) {
    __shared__ float sA[BM * LDP];
    __shared__ float sB[BN * LDP];
    __shared__ int   rlab[BM];
    __shared__ int   clab[BN];
    __shared__ float mneg[BM];

    const int row0 = blockIdx.y * BM;
    const int col0 = blockIdx.x * BN;
    const int tid  = threadIdx.x;

    rlab[tid] = labels[row0 + tid];
    mneg[tid] = maxneg[row0 + tid];
    if (tid < BN) clab[tid] = labels[col0 + tid];
    stage_tiles(E, row0, col0, tid, sA, sB);
    __syncthreads();

    const int wid  = tid >> 5;
    const int lane = tid & 31;
    const int half = lane >> 4;
    const int n16  = lane & 15;

    v8f acc[2][4];
    wave_gemm_lds(sA, sB, wid, lane, acc);

#pragma unroll
    for (int rt = 0; rt < 2; ++rt) {
#pragma unroll
        for (int r = 0; r < 8; ++r) {
            const int   rloc = 32 * wid + 16 * rt + r + 8 * half;
            const int   rl   = rlab[rloc];
            const float mn   = mneg[rloc];
            float s = 0.0f;
#pragma unroll
            for (int t = 0; t < 4; ++t) {
                if (rl == clab[16 * t + n16]) {
                    float v = acc[rt][t][r];
                    // sigmoid(mn - v) = 1 / (1 + e^{v - mn})
                    s += 1.0f / (1.0f + __expf(v - mn));
                }
            }
#pragma unroll
            for (int off = 1; off < 16; off <<= 1)
                s += __shfl_xor(s, off, 32);
            if (n16 == 0) atomicAdd(&possum[row0 + rloc], s);
        }
    }
}

__global__ void rk_finalize_kernel(const int* __restrict__ labels,
                                   const float* __restrict__ possum,
                                   const int* __restrict__ cnt,
                                   float* __restrict__ out) {
    int i = blockIdx.x * blockDim.x + threadIdx.x;
    float c = 0.0f;
    if (i < B_SIZE) {
        float mean = possum[i] / (float)cnt[labels[i]];
        c = (3.0f / (float)B_SIZE) * (1.0f - mean);   // len(K_VALUES) == 3
    }
#pragma unroll
    for (int off = 1; off < 32; off <<= 1)
        c += __shfl_xor(c, off, 32);
    if ((threadIdx.x & 31) == 0) atomicAdd(out, c);
}

extern "C" void kernel_launch(void* const* d_in, const int* in_sizes, int n_in,
                              void* d_out, int out_size, void* d_ws, size_t ws_size,
                              hipStream_t stream) {
    const float* E      = (const float*)d_in[0];
    const int*   labels = (const int*)d_in[1];
    float*       out    = (float*)d_out;

    float* maxneg = (float*)d_ws;                       // 8192 floats
    float* possum = maxneg + B_SIZE;                    // 8192 floats
    int*   cnt    = (int*)(possum + B_SIZE);            // 1000 ints

    rk_init_kernel<<<(B_SIZE + 255) / 256, 256, 0, stream>>>(maxneg, possum, cnt, out);
    rk_hist_kernel<<<(B_SIZE + 255) / 256, 256, 0, stream>>>(labels, cnt);

    dim3 grid(B_SIZE / BN, B_SIZE / BM);   // 128 x 64 tiles
    dim3 block(128);                       // 4 wave32s
    rk_maxneg_kernel<<<grid, block, 0, stream>>>(E, labels, maxneg);
    rk_possum_kernel<<<grid, block, 0, stream>>>(E, labels, maxneg, possum);

    rk_finalize_kernel<<<(B_SIZE + 255) / 256, 256, 0, stream>>>(labels, possum, cnt, out);
}